// SimpleGCNLayer_17025250361862
// MI455X (gfx1250) — compile-verified
//
#include <hip/hip_runtime.h>

#define D 128  // D_IN == D_OUT == 128

typedef __attribute__((ext_vector_type(2))) float v2f;
typedef __attribute__((ext_vector_type(8))) float v8f;

// ---------------------------------------------------------------------------
// Kernel 1: msg[n, o] = sum_k x[n,k] * W[o,k] + b[o]
// One wave32 computes one 16x16 output tile via V_WMMA_F32_16X16X4_F32.
// blockIdx.x = M-tile (16 nodes), wave-in-block (8 waves) = N-tile (16 feats).
// ---------------------------------------------------------------------------
__global__ void __launch_bounds__(256)
gcn_gemm_wmma(const float* __restrict__ x, const float* __restrict__ W,
              const float* __restrict__ bias, float* __restrict__ out) {
  const int lane = threadIdx.x & 31;
  const int wave = threadIdx.x >> 5;   // 0..7 -> N tile
  const int m0   = blockIdx.x << 4;    // node row base
  const int n0   = wave << 4;          // output-feature base
  const int half = lane >> 4;          // 0: lanes 0-15, 1: lanes 16-31
  const int l16  = lane & 15;

  // A frag: A[M=m0+l16][K = k0 + half*2 + {0,1}]  (ISA 16x4 f32 layout)
  const float* xp = x + (m0 + l16) * D + half * 2;
  // B frag: B[K][N=n0+l16] = W[n0+l16][K], same per-lane K split as A
  const float* wp = W + (n0 + l16) * D + half * 2;

  v8f acc = {};
#pragma unroll
  for (int k = 0; k < D; k += 4) {
    v2f a = *(const v2f*)(xp + k);
    v2f bfrag = *(const v2f*)(wp + k);
    // 8 args: (neg_a, A, neg_b, B, c_mod, C, reuse_a, reuse_b)
    acc = __builtin_amdgcn_wmma_f32_16x16x4_f32(false, a, false, bfrag,
                                                (short)0, acc, false, false);
  }

  // C/D layout: VGPR r -> row m0 + r + 8*half, col n0 + l16
  const float bv = bias[n0 + l16];
  float* op = out + (m0 + half * 8) * D + (n0 + l16);
#pragma unroll
  for (int r = 0; r < 8; ++r)
    op[r * D] = acc[r] + bv;
}

// Scalar fallback for a non-multiple-of-16 node tail (unused for N=20000).
__global__ void gcn_gemm_tail(const float* __restrict__ x, const float* __restrict__ W,
                              const float* __restrict__ bias, float* __restrict__ out,
                              int row0, int nrows) {
  int idx = blockIdx.x * blockDim.x + threadIdx.x;
  int r = idx / D, c = idx % D;
  if (r >= nrows) return;
  int row = row0 + r;
  float acc = bias[c];
  for (int k = 0; k < D; ++k) acc += x[row * D + k] * W[c * D + k];
  out[row * D + c] = acc;
}

// ---------------------------------------------------------------------------
// Kernel 2: zero the output (harness poisons d_out with 0xAA).
// ---------------------------------------------------------------------------
__global__ void zero_f4(float4* __restrict__ out, int n4) {
  int i = blockIdx.x * blockDim.x + threadIdx.x;
  if (i < n4) out[i] = make_float4(0.f, 0.f, 0.f, 0.f);
}

// ---------------------------------------------------------------------------
// Kernel 3: out[dst[e]] += msg[src[e]]  — one wave per edge, float4 per lane.
// msg (10.2 MB) is L2-resident; relaxed agent-scope fadd resolves at L2.
// ---------------------------------------------------------------------------
__global__ void __launch_bounds__(256)
scatter_add(const float* __restrict__ msg, const int* __restrict__ srcv,
            const int* __restrict__ dstv, float* __restrict__ out, int num_edges) {
  int gid = blockIdx.x * blockDim.x + threadIdx.x;
  int e = gid >> 5;  // wave-uniform edge id
  if (e >= num_edges) return;
  int lane = gid & 31;
  int s = srcv[e];
  int d = dstv[e];
  const float4 v = *(const float4*)(msg + s * D + lane * 4);
  float* o = out + d * D + lane * 4;
  __hip_atomic_fetch_add(o + 0, v.x, __ATOMIC_RELAXED, __HIP_MEMORY_SCOPE_AGENT);
  __hip_atomic_fetch_add(o + 1, v.y, __ATOMIC_RELAXED, __HIP_MEMORY_SCOPE_AGENT);
  __hip_atomic_fetch_add(o + 2, v.z, __ATOMIC_RELAXED, __HIP_MEMORY_SCOPE_AGENT);
  __hip_atomic_fetch_add(o + 3, v.w, __ATOMIC_RELAXED, __HIP_MEMORY_SCOPE_AGENT);
}

extern "C" void kernel_launch(void* const* d_in, const int* in_sizes, int n_in,
                              void* d_out, int out_size, void* d_ws, size_t ws_size,
                              hipStream_t stream) {
  const float* x    = (const float*)d_in[0];
  const int*   edge = (const int*)d_in[1];   // [2, E] row-major: src then dst
  const float* W    = (const float*)d_in[2];
  const float* b    = (const float*)d_in[3];
  float* out = (float*)d_out;
  float* msg = (float*)d_ws;                 // [num_nodes, D] intermediate

  const int num_nodes = in_sizes[0] / D;     // 20000
  const int num_edges = in_sizes[1] / 2;     // 640000

  // 1) msg = x @ W^T + b  (WMMA fp32)
  const int mtiles = num_nodes / 16;
  if (mtiles > 0)
    gcn_gemm_wmma<<<mtiles, 256, 0, stream>>>(x, W, b, msg);
  const int tail = num_nodes - mtiles * 16;
  if (tail > 0) {
    int work = tail * D;
    gcn_gemm_tail<<<(work + 255) / 256, 256, 0, stream>>>(x, W, b, msg,
                                                          mtiles * 16, tail);
  }

  // 2) out = 0
  const int n4 = (num_nodes * D) / 4;
  zero_f4<<<(n4 + 255) / 256, 256, 0, stream>>>((float4*)out, n4);

  // 3) out[dst] += msg[src]
  const long long threads = (long long)num_edges * 32;
  const int blocks = (int)((threads + 255) / 256);
  scatter_add<<<blocks, 256, 0, stream>>>(msg, edge, edge + num_edges, out,
                                          num_edges);
}